// GCN_Sparse_Memory_Policy_SelectNode_5_66726611911067
// MI455X (gfx1250) — compile-verified
//
#include <hip/hip_runtime.h>
#include <math.h>

#define N_NODES 100000
#define N_EDGES 3200000
#define NF      128

typedef float v2f __attribute__((ext_vector_type(2)));
typedef float v8f __attribute__((ext_vector_type(8)));

// ---------------------------------------------------------------------------
// Dense GEMM: S[N,128] = (relu_in ? relu(X) : X) @ W   using V_WMMA_F32_16X16X4_F32
// block = 256 threads = 8 waves; blockIdx.x selects a 16-row block,
// wave id selects the 16-column tile (8 tiles cover 128 columns).
// ---------------------------------------------------------------------------
__global__ __launch_bounds__(256) void gemm_wmma_f32(
    const float* __restrict__ X, const float* __restrict__ W,
    float* __restrict__ S, int relu_in)
{
    const int wave = threadIdx.x >> 5;          // 0..7  -> column tile
    const int lane = threadIdx.x & 31;
    const int row0 = blockIdx.x * 16;
    const int col0 = wave * 16;
    const int m    = lane & 15;                 // M (for A) / N (for B/C)
    const int kp   = (lane >> 4) << 1;          // 0 for lanes 0-15, 2 for 16-31

    const float* __restrict__ xrow = X + (size_t)(row0 + m) * NF;

    v8f acc = {};
#pragma unroll
    for (int k = 0; k < NF; k += 4) {
        // A tile 16x4 : lane m holds K = kp, kp+1 (consecutive -> b64 load)
        float2 av = *(const float2*)(xrow + k + kp);
        if (relu_in) { av.x = fmaxf(av.x, 0.0f); av.y = fmaxf(av.y, 0.0f); }
        v2f a; a.x = av.x; a.y = av.y;
        // B tile 4x16 : lane n=m holds rows K = kp, kp+1 of W
        v2f b;
        b.x = W[(size_t)(k + kp    ) * NF + col0 + m];
        b.y = W[(size_t)(k + kp + 1) * NF + col0 + m];
        acc = __builtin_amdgcn_wmma_f32_16x16x4_f32(
            /*neg_a=*/false, a, /*neg_b=*/false, b,
            /*c_mod=*/(short)0, acc, /*reuse_a=*/false, /*reuse_b=*/false);
    }

    // D layout: lanes 0-15 -> rows 0..7 in acc[0..7]; lanes 16-31 -> rows 8..15
    const int mbase = (lane >> 4) * 8;
    float* __restrict__ srow = S + (size_t)(row0 + mbase) * NF + col0 + m;
#pragma unroll
    for (int v = 0; v < 8; ++v)
        srow[(size_t)v * NF] = acc[v];
}

// ---------------------------------------------------------------------------
// Y[n, f] = b[f]   (initializes the aggregation target with the bias)
// ---------------------------------------------------------------------------
__global__ void bias_init_128(float* __restrict__ Y, const float* __restrict__ b)
{
    size_t i = (size_t)blockIdx.x * blockDim.x + threadIdx.x;
    if (i < (size_t)N_NODES * NF) Y[i] = b[i & (NF - 1)];
}

__global__ void scalar_init(float* __restrict__ z, const float* __restrict__ b)
{
    int i = blockIdx.x * blockDim.x + threadIdx.x;
    if (i < N_NODES) z[i] = b[0];
}

// ---------------------------------------------------------------------------
// Sparse aggregation: Y[row[e], :] += w[e] * S[col[e], :]
// One wave per edge; each lane owns 4 consecutive floats (float4 gather,
// 4x global_atomic_add_f32 scatter). All hot data lives in L2 (51.2 MB << 192 MB).
// ---------------------------------------------------------------------------
__global__ __launch_bounds__(256) void edge_scatter_128(
    const int* __restrict__ erow, const int* __restrict__ ecol,
    const float* __restrict__ ew, const float* __restrict__ S,
    float* __restrict__ Y)
{
    const int waveId = blockIdx.x * 8 + (threadIdx.x >> 5);
    const int lane   = threadIdx.x & 31;
    if (waveId >= N_EDGES) return;
    const int   r  = erow[waveId];
    const int   c  = ecol[waveId];
    const float wt = ew[waveId];

    const float4 sv = *(const float4*)(S + (size_t)c * NF + lane * 4);
    float* __restrict__ dst = Y + (size_t)r * NF + lane * 4;
    atomicAdd(dst + 0, wt * sv.x);
    atomicAdd(dst + 1, wt * sv.y);
    atomicAdd(dst + 2, wt * sv.z);
    atomicAdd(dst + 3, wt * sv.w);
}

// Layer-5 support: s5[n] = dot(relu(X[n,:]), W5[:,0]) ; wave per node.
__global__ __launch_bounds__(256) void dot128_relu(
    const float* __restrict__ X, const float* __restrict__ W5,
    float* __restrict__ s5)
{
    const int waveId = blockIdx.x * 8 + (threadIdx.x >> 5);
    const int lane   = threadIdx.x & 31;
    if (waveId >= N_NODES) return;
    float4 xv = *(const float4*)(X + (size_t)waveId * NF + lane * 4);
    const float4 wv = *(const float4*)(W5 + lane * 4);
    xv.x = fmaxf(xv.x, 0.0f); xv.y = fmaxf(xv.y, 0.0f);
    xv.z = fmaxf(xv.z, 0.0f); xv.w = fmaxf(xv.w, 0.0f);
    float s = xv.x * wv.x + xv.y * wv.y + xv.z * wv.z + xv.w * wv.w;
#pragma unroll
    for (int off = 16; off > 0; off >>= 1)
        s += __shfl_xor(s, off, 32);
    if (lane == 0) s5[waveId] = s;
}

// Scalar sparse aggregation for the last layer: z[row[e]] += w[e]*s5[col[e]]
__global__ void edge_scatter_1(
    const int* __restrict__ erow, const int* __restrict__ ecol,
    const float* __restrict__ ew, const float* __restrict__ s5,
    float* __restrict__ z)
{
    int e = blockIdx.x * blockDim.x + threadIdx.x;
    if (e >= N_EDGES) return;
    atomicAdd(&z[erow[e]], ew[e] * s5[ecol[e]]);
}

// ---------------------------------------------------------------------------
// log_softmax over 100000 elements: two-stage max, two-stage sum-exp, write.
// ---------------------------------------------------------------------------
__global__ void reduce_max_partial(const float* __restrict__ z, float* __restrict__ pmax)
{
    __shared__ float sm[256];
    const int tid = threadIdx.x;
    const int i   = blockIdx.x * 256 + tid;
    sm[tid] = (i < N_NODES) ? z[i] : -INFINITY;
    __syncthreads();
    for (int s = 128; s > 0; s >>= 1) {
        if (tid < s) sm[tid] = fmaxf(sm[tid], sm[tid + s]);
        __syncthreads();
    }
    if (tid == 0) pmax[blockIdx.x] = sm[0];
}

__global__ void reduce_max_final(const float* __restrict__ pmax, int np, float* __restrict__ scal)
{
    __shared__ float sm[512];
    const int tid = threadIdx.x;
    sm[tid] = (tid < np) ? pmax[tid] : -INFINITY;
    __syncthreads();
    for (int s = 256; s > 0; s >>= 1) {
        if (tid < s) sm[tid] = fmaxf(sm[tid], sm[tid + s]);
        __syncthreads();
    }
    if (tid == 0) scal[0] = sm[0];
}

__global__ void reduce_sumexp_partial(const float* __restrict__ z,
                                      const float* __restrict__ scal,
                                      float* __restrict__ psum)
{
    __shared__ float sm[256];
    const int tid = threadIdx.x;
    const int i   = blockIdx.x * 256 + tid;
    const float mx = scal[0];
    sm[tid] = (i < N_NODES) ? __expf(z[i] - mx) : 0.0f;
    __syncthreads();
    for (int s = 128; s > 0; s >>= 1) {
        if (tid < s) sm[tid] += sm[tid + s];
        __syncthreads();
    }
    if (tid == 0) psum[blockIdx.x] = sm[0];
}

__global__ void reduce_sum_final(const float* __restrict__ psum, int np, float* __restrict__ scal)
{
    __shared__ float sm[512];
    const int tid = threadIdx.x;
    sm[tid] = (tid < np) ? psum[tid] : 0.0f;
    __syncthreads();
    for (int s = 256; s > 0; s >>= 1) {
        if (tid < s) sm[tid] += sm[tid + s];
        __syncthreads();
    }
    if (tid == 0) scal[1] = scal[0] + logf(sm[0]);   // logsumexp
}

__global__ void logsoftmax_write(const float* __restrict__ z,
                                 const float* __restrict__ scal,
                                 float* __restrict__ out)
{
    int i = blockIdx.x * blockDim.x + threadIdx.x;
    if (i < N_NODES) out[i] = z[i] - scal[1];
}

// ---------------------------------------------------------------------------
// Launch
// ---------------------------------------------------------------------------
extern "C" void kernel_launch(void* const* d_in, const int* in_sizes, int n_in,
                              void* d_out, int out_size, void* d_ws, size_t ws_size,
                              hipStream_t stream)
{
    const float* features = (const float*)d_in[0];
    const int*   erow     = (const int*)  d_in[1];
    const int*   ecol     = (const int*)  d_in[2];
    const float* ew       = (const float*)d_in[3];
    const float* W[5] = { (const float*)d_in[4], (const float*)d_in[6],
                          (const float*)d_in[8], (const float*)d_in[10],
                          (const float*)d_in[12] };
    const float* B[5] = { (const float*)d_in[5], (const float*)d_in[7],
                          (const float*)d_in[9], (const float*)d_in[11],
                          (const float*)d_in[13] };
    float* out = (float*)d_out;

    const size_t BIG = (size_t)N_NODES * NF * sizeof(float);   // 51.2 MB
    char* ws = (char*)d_ws;
    float* bufA = (float*)(ws);             // support / s5
    float* bufB = (float*)(ws + BIG);       // activations ping
    float* bufC = (float*)(ws + 2 * BIG);   // activations pong
    float* scal = (float*)(ws + 3 * BIG);   // [0]=max, [1]=lse
    float* pmax = scal + 8;                 // 512 partials
    float* psum = pmax + 512;

    const int gemmGrid   = N_NODES / 16;                 // 6250
    const int initGrid   = (int)(((size_t)N_NODES * NF + 255) / 256);
    const int edgeGrid   = (N_EDGES + 7) / 8;            // 8 waves / block
    const int nodeWGrid  = (N_NODES + 7) / 8;
    const int nodeGrid   = (N_NODES + 255) / 256;        // 391
    const int edge1Grid  = (N_EDGES + 255) / 256;

    // ---- layers 1..4:  S = act(X)@W ; Y = b ; Y += A*S ----
    const float* Xcur = features;
    float* Ys[4] = { bufB, bufC, bufB, bufC };
    for (int l = 0; l < 4; ++l) {
        gemm_wmma_f32<<<gemmGrid, 256, 0, stream>>>(Xcur, W[l], bufA, l > 0);
        bias_init_128<<<initGrid, 256, 0, stream>>>(Ys[l], B[l]);
        edge_scatter_128<<<edgeGrid, 256, 0, stream>>>(erow, ecol, ew, bufA, Ys[l]);
        Xcur = Ys[l];
    }

    // ---- layer 5: 128 -> 1 ----
    dot128_relu<<<nodeWGrid, 256, 0, stream>>>(bufC, W[4], bufA);   // s5 in bufA
    scalar_init<<<nodeGrid, 256, 0, stream>>>(bufB, B[4]);          // z in bufB
    edge_scatter_1<<<edge1Grid, 256, 0, stream>>>(erow, ecol, ew, bufA, bufB);

    // ---- log_softmax ----
    reduce_max_partial<<<nodeGrid, 256, 0, stream>>>(bufB, pmax);
    reduce_max_final<<<1, 512, 0, stream>>>(pmax, nodeGrid, scal);
    reduce_sumexp_partial<<<nodeGrid, 256, 0, stream>>>(bufB, scal, psum);
    reduce_sum_final<<<1, 512, 0, stream>>>(psum, nodeGrid, scal);
    logsoftmax_write<<<nodeGrid, 256, 0, stream>>>(bufB, scal, out);
}